// CTCLossLayer_24154896072978
// MI455X (gfx1250) — compile-verified
//
#include <hip/hip_runtime.h>
#include <math.h>

#define NEGV (-1e30f)
#define EPSV (1e-7f)

// logaddexp: max + log(1 + exp(-|a-b|)). Arg of log is in (1,2] -> __logf is safe.
// With -1e30 sentinels: exp(-huge)=0 -> returns max exactly (matches reference fp32).
__device__ __forceinline__ float lae(float a, float b) {
    return fmaxf(a, b) + __logf(1.0f + __expf(-fabsf(a - b)));
}

// One wave32 per batch element. Lane k owns CTC states 4k..4k+3 (lane 31 also
// owns state 128). Alpha lives in registers; the only cross-lane dependency per
// timestep is prev-lane old[3], fetched with one __shfl_up. y_pred rows are
// streamed global->LDS through a depth-8 ring of CDNA5 async-to-LDS DMA loads,
// self-synchronized with ASYNCcnt (no barriers anywhere). Depth 8 gives ~8
// compute-steps of latency slack, needed since occupancy is ~1 wave/SIMD.
__global__ __launch_bounds__(32)
void ctc_loss_gfx1250_kernel(const float* __restrict__ y_pred,
                             const int*  __restrict__ labels,
                             const int*  __restrict__ input_length,
                             const int*  __restrict__ label_length,
                             float*      __restrict__ out,
                             int T, int L)
{
    constexpr int C    = 128;          // classes: one row = 512B = 32 lanes x b128
    constexpr int ROWB = C * 4;
    constexpr int D    = 8;            // DMA ring depth (power of two)
    const int b     = blockIdx.x;
    const int lane  = threadIdx.x;     // 0..31
    const int S     = 2 * L + 1;       // 129
    const int blank = C - 1;

    __shared__ float ybuf[D][C];       // row ring (4 KB)

    int t_idx = input_length[b] - 1;
    t_idx = max(0, min(t_idx, T - 1));
    const int s_end = 2 * label_length[b];     // provably even

    // Per-lane state metadata. s = 4*lane + j ; parity(s) == parity(j).
    int  lab[5];  bool skip[5]; bool valid[5]; bool capE[5], capM[5];
    #pragma unroll
    for (int j = 0; j < 5; ++j) {
        const int s = lane * 4 + j;
        valid[j] = (j < 4) ? true : ((lane == 31) && (s < S));
        lab[j]  = blank;
        skip[j] = false;
        if ((j & 1) && valid[j]) {               // odd states carry a label
            lab[j] = labels[(size_t)b * L + (s >> 1)];
            if (s >= 2)
                skip[j] = (lab[j] != labels[(size_t)b * L + (s >> 1) - 1]);
        }
        capE[j] = valid[j] && (s == s_end);      // folds: only even j possible
        capM[j] = valid[j] && (s == s_end - 1);  // folds: only odd j possible
    }

    const uint64_t gbase = (uint64_t)(uintptr_t)(y_pred + (size_t)b * T * C);
    const uint32_t lbase = (uint32_t)(uintptr_t)(&ybuf[0][0]);  // low 32b = LDS offset
    const uint64_t glane = gbase + (uint64_t)lane * 16u;
    const uint32_t llane = lbase + (uint32_t)lane * 16u;

    // Prime rows 0..D-1 (T=512 >> D here); ensure row 0 has landed.
    #pragma unroll
    for (int r = 0; r < D; ++r) {
        uint64_t ga = glane + (uint64_t)r * ROWB;
        uint32_t la = llane + (uint32_t)r * ROWB;
        asm volatile("global_load_async_to_lds_b128 %0, %1, off"
                     :: "v"(la), "v"(ga) : "memory");
    }
    asm volatile("s_wait_asynccnt 0x7" ::: "memory");   // >=1 done -> row 0 ready

    // t = 0 : alpha0[s] = (s < 2) ? log(y[lab]+eps) : NEG
    float a[5];
    float capEv = NEGV, capMv = NEGV;
    {
        const float* row = ybuf[0];
        float lpB = __logf(row[blank] + EPSV);
        #pragma unroll
        for (int j = 0; j < 5; ++j) {
            const int s = lane * 4 + j;
            float lp = (j & 1) ? __logf(row[lab[j]] + EPSV) : lpB;
            a[j] = (s < 2) ? lp : NEGV;
        }
        if (t_idx == 0) {
            #pragma unroll
            for (int j = 0; j < 5; ++j) {
                if (capE[j]) capEv = a[j];
                if (capM[j]) capMv = a[j];
            }
        }
        asm volatile("s_wait_dscnt 0x0" ::: "memory");   // row-0 reads done
        if (D < T) {                                     // prefetch row D -> buf 0
            asm volatile("global_load_async_to_lds_b128 %0, %1, off"
                         :: "v"(llane), "v"(glane + (uint64_t)D * ROWB) : "memory");
            asm volatile("s_wait_asynccnt 0x7" ::: "memory");  // row 1 resident
        } else {
            asm volatile("s_wait_asynccnt 0x0" ::: "memory");
        }
    }

    for (int t = 1; t < T; ++t) {
        const int buf = t & (D - 1);
        const float* row = ybuf[buf];

        // prev lane's old[3] == a[s-1] for this lane's j=0 (and a[s-2] for j=1)
        float nb1 = __shfl_up(a[3], 1, 32);
        if (lane == 0) nb1 = NEGV;

        const float am1[5] = { nb1,  a[0], a[1], a[2], a[3] };
        const float am2[5] = { NEGV, nb1,  NEGV, a[1], NEGV };  // used only by odd j

        float lpB = __logf(row[blank] + EPSV);
        float nw[5];
        #pragma unroll
        for (int j = 0; j < 5; ++j) {
            float l = lae(a[j], am1[j]);
            float lp;
            if (j & 1) {                          // odd: label state, skip path live
                float a3 = skip[j] ? am2[j] : NEGV;
                l  = lae(l, a3);
                lp = __logf(row[lab[j]] + EPSV);
            } else {                              // even: blank, skip==false =>
                lp = lpB;                         // lae(l, NEG) == l exactly
            }
            nw[j] = l + lp;
        }

        if (t == t_idx) {
            #pragma unroll
            for (int j = 0; j < 5; ++j) {
                if (capE[j]) capEv = nw[j];
                if (capM[j]) capMv = nw[j];
            }
        }
        #pragma unroll
        for (int j = 0; j < 5; ++j) a[j] = nw[j];

        // All row-t LDS gathers retired -> safe to overwrite buf with row t+D.
        asm volatile("s_wait_dscnt 0x0" ::: "memory");
        const int tf = t + D;
        if (tf < T) {
            uint64_t ga = glane + (uint64_t)tf * ROWB;
            uint32_t la = llane + (uint32_t)buf * ROWB;
            asm volatile("global_load_async_to_lds_b128 %0, %1, off"
                         :: "v"(la), "v"(ga) : "memory");
            asm volatile("s_wait_asynccnt 0x7" ::: "memory");  // row t+1 resident
        } else {
            asm volatile("s_wait_asynccnt 0x0" ::: "memory");  // tail drain
        }
    }

    // Gather the two captured alphas from their owner lanes and emit the loss.
    int laneE = (s_end >= 128) ? 31 : (s_end >> 2);           // s_end==128 -> lane31 j=4
    int laneM = max(0, (s_end - 1) >> 2);
    float aE = __shfl(capEv, laneE, 32);
    float aM = __shfl(capMv, laneM, 32);
    if (lane == 0) out[b] = -lae(aE, aM);
}

extern "C" void kernel_launch(void* const* d_in, const int* in_sizes, int n_in,
                              void* d_out, int out_size, void* d_ws, size_t ws_size,
                              hipStream_t stream) {
    const float* y_pred       = (const float*)d_in[0];
    const int*   labels       = (const int*)d_in[1];
    const int*   input_length = (const int*)d_in[2];
    const int*   label_length = (const int*)d_in[3];
    float*       out          = (float*)d_out;

    const int B = in_sizes[2];                 // input_length: (B,1)
    const int L = in_sizes[1] / B;             // labels: (B,L)
    const int C = 128;
    const int T = in_sizes[0] / (B * C);       // y_pred: (B,T,C)

    ctc_loss_gfx1250_kernel<<<B, 32, 0, stream>>>(
        y_pred, labels, input_length, label_length, out, T, L);
}